// upconv_Embedding_43774306681497
// MI455X (gfx1250) — compile-verified
//
#include <hip/hip_runtime.h>
#include <hip/hip_bf16.h>

typedef __attribute__((ext_vector_type(2))) float v2f;
typedef __attribute__((ext_vector_type(4))) float v4f;
typedef __attribute__((ext_vector_type(8))) float v8f;

#define H_DIM 16
#define KC_DIM 16
#define OC 128
#define IC 128
#define KW 3
#define HID 256
#define ZD 64
#define NPDE 1024
#define M_GEMM 256            // rows of hidden = H*KC
#define N_GEMM 49152          // OC*IC*KW
#define K_GEMM 256            // HID
#define NTILES (N_GEMM / 16)  // 3072
#define OUT_W2D (2048LL * 2048 * 3 * 3)  // 37748736

// ---------------- kernel 1: pde_feat = mean(pde_inputs,0) @ U1 + b1 ----------
__global__ __launch_bounds__(256) void k_pde_feat(const float* __restrict__ pde,
                                                  const float* __restrict__ U1,
                                                  const float* __restrict__ b1,
                                                  float* __restrict__ feat) {
  __shared__ float colmean[ZD];
  int t = threadIdx.x;
  if (t < ZD) {
    float s = 0.f;
    for (int n = 0; n < NPDE; ++n) s += pde[n * ZD + t];
    colmean[t] = s * (1.0f / (float)NPDE);
  }
  __syncthreads();
  float s = b1[t];
  for (int d = 0; d < ZD; ++d) s += colmean[d] * U1[d * HID + t];
  feat[t] = s;
}

// ---------------- kernel 2: hidden = tanh(Z @ W1 + feat) ---------------------
__global__ __launch_bounds__(256) void k_hidden(const float* __restrict__ Z,
                                                const float* __restrict__ W1,
                                                const float* __restrict__ feat,
                                                float* __restrict__ hidden) {
  int r = blockIdx.x;        // 0..255
  int c = threadIdx.x;       // 0..255
  __shared__ float zrow[ZD];
  if (c < ZD) zrow[c] = Z[r * ZD + c];
  __syncthreads();
  float s = feat[c];
  #pragma unroll 8
  for (int d = 0; d < ZD; ++d) s += zrow[d] * W1[d * HID + c];
  hidden[r * HID + c] = tanhf(s);
}

// ---------------- kernel 3: bb = z_bias + unet_1d_bias -----------------------
__global__ void k_bias(const float* __restrict__ zb, const float* __restrict__ ub,
                       float* __restrict__ out) {
  int t = blockIdx.x * blockDim.x + threadIdx.x;
  if (t < 2048) out[OUT_W2D + t] = zb[t] + ub[t];
}

// ---------------- kernel 4: fused GEMM (hidden @ W2 + b2) + outer-product ----
// Wave: 4 M-tiles x 1 N-tile, f32 WMMA 16x16x4, K = 256.
// Epilogue: each 16x16 w-tile row expands to 48 contiguous output floats
// (k,l interleaved), multiplied by unet_1d_weights, stored as 12 x b128.
__global__ __launch_bounds__(256) void k_main(const float* __restrict__ hidden,
                                              const float* __restrict__ W2,
                                              const float* __restrict__ b2,
                                              const float* __restrict__ unet,
                                              float* __restrict__ out) {
  __shared__ float lds[8][64 * 17];  // per-wave staging, bank-padded
  const int lane   = threadIdx.x & 31;
  const int wib    = threadIdx.x >> 5;
  const int w      = blockIdx.x * 8 + wib;   // 0..12287
  const int tn     = w >> 2;                 // N-tile 0..3071
  const int mb     = w & 3;                  // M-block 0..3 (4 tiles each)
  const int laneLo = lane & 15;
  const int laneHi = lane >> 4;

  const int c0 = tn * 16;
  const int r0 = mb * 64;

  v8f acc0 = {}, acc1 = {}, acc2 = {}, acc3 = {};

  // A: lane holds hidden[r0 + mt*16 + laneLo][kk + laneHi*2 .. +1]  (float2)
  // B: lane holds W2[kk + laneHi*2 .. +1][c0 + laneLo]
  const float* Abase = hidden + (r0 + laneLo) * K_GEMM + laneHi * 2;
  const float* Bbase = W2 + (size_t)(laneHi * 2) * N_GEMM + c0 + laneLo;

  for (int kk = 0; kk < K_GEMM; kk += 4) {
    v2f bfrag;
    bfrag.x = Bbase[(size_t)kk * N_GEMM];
    bfrag.y = Bbase[(size_t)(kk + 1) * N_GEMM];
    v2f a0 = *(const v2f*)(Abase + 0 * 16 * K_GEMM + kk);
    v2f a1 = *(const v2f*)(Abase + 1 * 16 * K_GEMM + kk);
    v2f a2 = *(const v2f*)(Abase + 2 * 16 * K_GEMM + kk);
    v2f a3 = *(const v2f*)(Abase + 3 * 16 * K_GEMM + kk);
    acc0 = __builtin_amdgcn_wmma_f32_16x16x4_f32(false, a0, false, bfrag,
                                                 (short)0, acc0, false, false);
    acc1 = __builtin_amdgcn_wmma_f32_16x16x4_f32(false, a1, false, bfrag,
                                                 (short)0, acc1, false, false);
    acc2 = __builtin_amdgcn_wmma_f32_16x16x4_f32(false, a2, false, bfrag,
                                                 (short)0, acc2, false, false);
    acc3 = __builtin_amdgcn_wmma_f32_16x16x4_f32(false, a3, false, bfrag,
                                                 (short)0, acc3, false, false);
  }

  // add b2 (column bias) and stage tiles to LDS: lds row = tile-local M, col = N
  const float bias2 = b2[c0 + laneLo];
  float* L = lds[wib];
  #pragma unroll
  for (int v = 0; v < 8; ++v) {
    const int mrow = v + laneHi * 8;
    L[(0  + mrow) * 17 + laneLo] = acc0[v] + bias2;
    L[(16 + mrow) * 17 + laneLo] = acc1[v] + bias2;
    L[(32 + mrow) * 17 + laneLo] = acc2[v] + bias2;
    L[(48 + mrow) * 17 + laneLo] = acc3[v] + bias2;
  }
  __syncthreads();

  // epilogue: c = oc*384 + s, s = s0 + j, j = tile col, ic = s/3, k = s%3
  // out[((o*2048+i)*3+k)*3+l] = w * unet[(o*2048+i)*3 + l]
  // per tile row v: 48 contiguous floats at o*18432 + v*1152 + 3*s0 + (3j+l)
  const int oc = tn / 24;
  const int s0 = (tn % 24) * 16;
  #pragma unroll
  for (int mt = 0; mt < 4; ++mt) {
    const int h = mb * 4 + mt;
    const int o = h * 128 + oc;
    const int obase = o * 18432 + 3 * s0;
    const int ubase = o * 6144;
    const float* Lt = L + mt * 16 * 17;
    #pragma unroll
    for (int it = 0; it < 6; ++it) {
      const int pair = it * 32 + lane;   // 0..191: (row v, quad g)
      const int v = pair / 12;
      const int g = pair % 12;
      const float* Crow = Lt + v * 17;
      const float* Urow = unet + ubase + v * 384;
      v4f val;
      #pragma unroll
      for (int e = 0; e < 4; ++e) {
        const int p  = 4 * g + e;       // 0..47 within row
        const int j  = p / 3;           // tile column
        const int l  = p - 3 * j;       // unet L index
        const int ic = (s0 + j) / 3;
        val[e] = Crow[j] * Urow[ic * 3 + l];
      }
      *(v4f*)(out + obase + v * 1152 + 4 * g) = val;
    }
  }
}

extern "C" void kernel_launch(void* const* d_in, const int* in_sizes, int n_in,
                              void* d_out, int out_size, void* d_ws, size_t ws_size,
                              hipStream_t stream) {
  (void)in_sizes; (void)n_in; (void)out_size; (void)ws_size;
  const float* Z      = (const float*)d_in[0];
  const float* z_bias = (const float*)d_in[1];
  const float* unet_w = (const float*)d_in[2];
  const float* unet_b = (const float*)d_in[3];
  const float* pde    = (const float*)d_in[4];
  const float* W1     = (const float*)d_in[5];
  const float* U1     = (const float*)d_in[6];
  const float* b1     = (const float*)d_in[7];
  const float* W2     = (const float*)d_in[8];
  const float* b2     = (const float*)d_in[9];
  float* out = (float*)d_out;

  float* feat   = (float*)d_ws;          // 256 floats
  float* hidden = feat + 256;            // 256*256 floats

  k_pde_feat<<<1, 256, 0, stream>>>(pde, U1, b1, feat);
  k_hidden<<<M_GEMM, 256, 0, stream>>>(Z, W1, feat, hidden);
  k_bias<<<8, 256, 0, stream>>>(z_bias, unet_b, out);
  // 12288 waves total, 8 waves (256 threads) per block
  k_main<<<(4 * NTILES) / 8, 256, 0, stream>>>(hidden, W2, b2, unet_w, out);
}